// MyACLIP_74990128988293
// MI455X (gfx1250) — compile-verified
//
#include <hip/hip_runtime.h>
#include <hip/hip_bf16.h>

typedef __attribute__((ext_vector_type(2))) float v2f;
typedef __attribute__((ext_vector_type(8))) float v8f;

#define BS 8192
#define NCLS 400
#define DIM 768
#define LDS_STRIDE 772   // 768 + 4: bank-conflict-free ds_load_b64, keeps 16B align

// ---------------------------------------------------------------------------
// GEMM: out[m][n] = sum_k E[m][k] * W[n][k] + b[n]
// Block = 256 threads (8 waves). Each wave owns a 16x16 output tile.
// Block covers 128 M-rows x 16 N-cols; W tile (16 x 768) cached in LDS.
// Uses V_WMMA_F32_16X16X4_F32 (full fp32 matrix path on CDNA5).
// ---------------------------------------------------------------------------
__global__ __launch_bounds__(256) void aclip_gemm_wmma(
    const float* __restrict__ E, const float* __restrict__ W,
    const float* __restrict__ bias, float* __restrict__ out) {
  __shared__ float s_w[16 * LDS_STRIDE];

  const int n0 = blockIdx.y * 16;
  const int tid = threadIdx.x;

  // Cooperative fill of the W tile: rows n0..n0+15, 768 floats each (b128).
  for (int i = tid; i < 16 * (DIM / 4); i += 256) {
    const int row = i / (DIM / 4);
    const int kk = (i - row * (DIM / 4)) * 4;
    const float4 v = *(const float4*)(W + (size_t)(n0 + row) * DIM + kk);
    *(float4*)(s_w + row * LDS_STRIDE + kk) = v;
  }
  __syncthreads();

  const int wave = tid >> 5;
  const int lane = tid & 31;
  const int m0 = blockIdx.x * 128 + wave * 16;
  const int lidx = lane & 15;              // M index for A, N index for B
  const int koff = (lane >> 4) << 1;       // lanes 16-31 carry K+2,K+3

  const float* arow = E + (size_t)(m0 + lidx) * DIM + koff;     // global (L2-resident)
  const float* brow = s_w + lidx * LDS_STRIDE + koff;           // LDS

  v8f acc = {};
#pragma unroll 4
  for (int k = 0; k < DIM; k += 4) {
    const v2f a = *(const v2f*)(arow + k);   // A[M=lidx][k+koff, k+koff+1]
    const v2f b = *(const v2f*)(brow + k);   // B[k+koff, k+koff+1][N=lidx]
    acc = __builtin_amdgcn_wmma_f32_16x16x4_f32(
        /*neg_a=*/false, a, /*neg_b=*/false, b,
        /*c_mod=*/(short)0, acc, /*reuse_a=*/false, /*reuse_b=*/false);
  }

  // C/D layout: VGPR r, lane l -> M = r + 8*(l>>4), N = l&15
  const float bn = bias[n0 + lidx];
  const int rowHi = (lane >> 4) * 8;
  const int col = n0 + lidx;
#pragma unroll
  for (int r = 0; r < 8; ++r) {
    out[(size_t)(m0 + rowHi + r) * NCLS + col] = acc[r] + bn;
  }
}

// ---------------------------------------------------------------------------
// Mask pipeline: one block per (sample, box). 2*BS blocks total.
// ---------------------------------------------------------------------------
__global__ __launch_bounds__(256) void aclip_mask(
    const float* __restrict__ attn, const int* __restrict__ pos,
    float* __restrict__ outMask) {
  __shared__ float s_attn[196];
  __shared__ float s_cm[196];
  __shared__ float s_a7[49];
  __shared__ float s_m7[49];

  const int task = blockIdx.x;
  const int box = task >> 13;     // 0: pos[:,0], 1: pos[:,1]
  const int s = task & (BS - 1);
  const int t = threadIdx.x;

  if (t < 196) s_attn[t] = attn[(size_t)s * 196 + t];

  const int* pb = pos + (size_t)s * 12;       // pos[s] is 3x4 int32
  const int* ob = pb + box * 4;
  const float eH = (float)pb[10], eW = (float)pb[11];
  const float top = (float)(ob[0] - pb[8]);
  const float left = (float)(ob[1] - pb[9]);
  const float oH = (float)ob[2], oW = (float)ob[3];
  __syncthreads();

  if (t < 196) {
    const int oy = t / 14, ox = t - oy * 14;
    const float ty = fmaxf(((float)oy + 0.5f) * oH / 14.0f - 0.5f, 0.0f);
    const float y0 = floorf(ty);
    const float y1 = fminf(y0 + 1.0f, oH - 1.0f);
    const float wy = ty - y0;
    const float tx = fmaxf(((float)ox + 0.5f) * oW / 14.0f - 0.5f, 0.0f);
    const float x0 = floorf(tx);
    const float x1 = fminf(x0 + 1.0f, oW - 1.0f);
    const float wx = tx - x0;
    const float Y0 = top + y0, Y1 = top + y1, X0 = left + x0, X1 = left + x1;

    auto bilin = [&](float Y, float X) -> float {
      const float sy = fmaxf((Y + 0.5f) * 14.0f / eH - 0.5f, 0.0f);
      const float sx = fmaxf((X + 0.5f) * 14.0f / eW - 0.5f, 0.0f);
      const float r0f = floorf(sy), c0f = floorf(sx);
      const int r0 = (int)r0f;
      const int r1 = min(r0 + 1, 13);
      const int c0 = (int)c0f;
      const int c1 = min(c0 + 1, 13);
      const float fy = sy - r0f, fx = sx - c0f;
      const float g00 = s_attn[r0 * 14 + c0];
      const float g01 = s_attn[r0 * 14 + c1];
      const float g10 = s_attn[r1 * 14 + c0];
      const float g11 = s_attn[r1 * 14 + c1];
      return g00 * (1.0f - fy) * (1.0f - fx) + g01 * (1.0f - fy) * fx +
             g10 * fy * (1.0f - fx) + g11 * fy * fx;
    };

    const float v00 = bilin(Y0, X0);
    const float v01 = bilin(Y0, X1);
    const float v10 = bilin(Y1, X0);
    const float v11 = bilin(Y1, X1);
    s_cm[t] = v00 * (1.0f - wy) * (1.0f - wx) + v01 * (1.0f - wy) * wx +
              v10 * wy * (1.0f - wx) + v11 * wy * wx;
  }
  __syncthreads();

  if (t < 49) {
    const int py = t / 7, px = t - py * 7;
    const int base = (2 * py) * 14 + 2 * px;
    s_a7[t] = ((s_cm[base] + s_cm[base + 1]) +
               (s_cm[base + 14] + s_cm[base + 15])) * 0.25f;
  }
  __syncthreads();

  if (t < 49) {
    // top_k(25) with lower-index tie-break; mask is True for NON-selected.
    const float v = s_a7[t];
    int rank = 0;
#pragma unroll
    for (int j = 0; j < 49; ++j) {
      const float u = s_a7[j];
      rank += (u > v) || (u == v && j < t);
    }
    s_m7[t] = (rank >= 25) ? 1.0f : 0.0f;
  }
  __syncthreads();

  if (t < 196) {
    const int oy = t / 14, ox = t - oy * 14;
    outMask[(size_t)task * 196 + t] = s_m7[(oy >> 1) * 7 + (ox >> 1)];
  }
}

extern "C" void kernel_launch(void* const* d_in, const int* in_sizes, int n_in,
                              void* d_out, int out_size, void* d_ws, size_t ws_size,
                              hipStream_t stream) {
  const float* attn = (const float*)d_in[0];
  const int* pos = (const int*)d_in[1];
  const float* E = (const float*)d_in[2];      // image_embed (2*BS, D)
  const float* W = (const float*)d_in[3];      // (NCLS, D)
  const float* bias = (const float*)d_in[4];   // (NCLS,)

  float* logits = (float*)d_out;                              // (2*BS, NCLS)
  float* mask = logits + (size_t)(2 * BS) * NCLS;             // (2*BS, 196)

  // GEMM: grid (M/128, NCLS/16) = (128, 25)
  aclip_gemm_wmma<<<dim3(128, 25), 256, 0, stream>>>(E, W, bias, logits);

  // Masks: one block per (sample, box); concat order = box-major.
  aclip_mask<<<2 * BS, 256, 0, stream>>>(attn, pos, mask);
}